// SelectiveSSM_21388937134755
// MI455X (gfx1250) — compile-verified
//
#include <hip/hip_runtime.h>
#include <hip/hip_bf16.h>

// ---------------------------------------------------------------------------
// Selective SSM, MI455X (gfx1250) strategy:
//   x in {0,1}  =>  per-channel coefficients have only two values (a0,0)/(a1,b1).
//   3-pass chunked parallel scan (L=64, NC=64, 2048 WGs) kills the T=4096
//   sequential dependency; per-chunk output GEMV (64x512 . 512) runs on the
//   matrix pipe via V_WMMA_F32_16X16X4_F32 from a bank-conflict-free LDS tile.
//   HBM traffic ~13 MB total (P,Q,H0 + x + y)  => ~0.6us at 23.3 TB/s; the
//   kernel is latency-bound, which the chunking + WMMA offload addresses.
// ---------------------------------------------------------------------------

#define B_BATCH 32
#define T_LEN   4096
#define D_DIM   512
#define L_CHUNK 64
#define NC      (T_LEN / L_CHUNK)   // 64 chunks
#define STRIDE  (D_DIM + 4)         // 516 floats: 516%64==4 -> conflict-free

typedef __attribute__((ext_vector_type(2))) float v2f;
typedef __attribute__((ext_vector_type(8))) float v8f;

__device__ __forceinline__ float softplus_f(float z) {
    return (z > 20.0f) ? z : log1pf(__expf(z));
}

// Per-channel two-state coefficients (x is binary).
__device__ __forceinline__ void coeffs(int dd,
                                       const float* __restrict__ A_log,
                                       const float* __restrict__ dw,
                                       const float* __restrict__ db,
                                       const float* __restrict__ Bw,
                                       const float* __restrict__ Bb,
                                       float& a0, float& a1, float& b1) {
    float A  = -__expf(A_log[dd]);
    float d0 = softplus_f(db[dd]);
    float d1 = softplus_f(dw[dd] + db[dd]);
    a0 = __expf(d0 * A);
    a1 = __expf(d1 * A);
    b1 = d1 * (Bw[dd] + Bb[dd]);      // B_bar for x=1 (x=0 gives exactly 0)
}

// ---- Pass 1: per-(batch,chunk) compose the affine map h_out = P*h_in + Q ----
__global__ __launch_bounds__(128) void ssm_compose(
    const float* __restrict__ x, const float* __restrict__ A_log,
    const float* __restrict__ dw, const float* __restrict__ db,
    const float* __restrict__ Bw, const float* __restrict__ Bb,
    float* __restrict__ P, float* __restrict__ Q) {
    __shared__ float sx[L_CHUNK];
    const int c = blockIdx.x, b = blockIdx.y, tid = threadIdx.x;
    if (tid < L_CHUNK) sx[tid] = x[b * T_LEN + c * L_CHUNK + tid];

    const int d0i = tid * 4;
    float a0[4], a1[4], b1[4];
    #pragma unroll
    for (int j = 0; j < 4; ++j)
        coeffs(d0i + j, A_log, dw, db, Bw, Bb, a0[j], a1[j], b1[j]);
    __syncthreads();

    float Pr[4] = {1.f, 1.f, 1.f, 1.f};
    float Qr[4] = {0.f, 0.f, 0.f, 0.f};
    #pragma unroll 4
    for (int t = 0; t < L_CHUNK; ++t) {
        const bool bit = sx[t] > 0.5f;
        #pragma unroll
        for (int j = 0; j < 4; ++j) {
            const float a  = bit ? a1[j] : a0[j];
            const float bb = bit ? b1[j] : 0.0f;
            Pr[j] = a * Pr[j];
            Qr[j] = fmaf(a, Qr[j], bb);
        }
    }
    const size_t base = ((size_t)(b * NC + c)) * D_DIM + d0i;
    *(float4*)(P + base) = make_float4(Pr[0], Pr[1], Pr[2], Pr[3]);
    *(float4*)(Q + base) = make_float4(Qr[0], Qr[1], Qr[2], Qr[3]);
}

// ---- Pass 2: scan chunk boundaries: 16384 independent (b,d) lanes, 64 steps ----
__global__ __launch_bounds__(256) void ssm_chunkscan(
    const float* __restrict__ P, const float* __restrict__ Q,
    float* __restrict__ H0) {
    const int idx = blockIdx.x * 256 + threadIdx.x;   // 0 .. B*D-1
    const int b = idx >> 9, dch = idx & (D_DIM - 1);
    float h = 0.0f;
    #pragma unroll 8
    for (int c = 0; c < NC; ++c) {
        const size_t off = ((size_t)(b * NC + c)) * D_DIM + dch;
        if (c + 2 < NC) {   // pipeline the chunk-coefficient stream through L2
            __builtin_prefetch(P + off + 2 * D_DIM, 0, 3);
            __builtin_prefetch(Q + off + 2 * D_DIM, 0, 3);
        }
        H0[off] = h;                      // h entering chunk c
        h = fmaf(P[off], h, Q[off]);
    }
}

// ---- Pass 3: replay chunk into LDS tile, then GEMV y = sigmoid(H.C + d) via WMMA ----
__global__ __launch_bounds__(128) void ssm_replay(
    const float* __restrict__ x, const float* __restrict__ A_log,
    const float* __restrict__ dw, const float* __restrict__ db,
    const float* __restrict__ Bw, const float* __restrict__ Bb,
    const float* __restrict__ Cvec, const float* __restrict__ dscal,
    const float* __restrict__ H0, float* __restrict__ y) {
    extern __shared__ float smem[];
    float* tile = smem;                        // [L_CHUNK][STRIDE]
    float* sx   = smem + L_CHUNK * STRIDE;     // [L_CHUNK]
    float* sC   = sx + L_CHUNK;                // [D_DIM]

    const int c = blockIdx.x, b = blockIdx.y, tid = threadIdx.x;
    const int d0i = tid * 4;

    if (tid < L_CHUNK) sx[tid] = x[b * T_LEN + c * L_CHUNK + tid];
    *(float4*)(sC + d0i) = *(const float4*)(Cvec + d0i);

    float a0[4], a1[4], b1[4];
    #pragma unroll
    for (int j = 0; j < 4; ++j)
        coeffs(d0i + j, A_log, dw, db, Bw, Bb, a0[j], a1[j], b1[j]);

    const size_t hbase = ((size_t)(b * NC + c)) * D_DIM + d0i;
    const float4 h4 = *(const float4*)(H0 + hbase);
    float h[4] = {h4.x, h4.y, h4.z, h4.w};
    __syncthreads();

    // Sequential replay within chunk (depth 64), channels fully parallel.
    #pragma unroll 4
    for (int t = 0; t < L_CHUNK; ++t) {
        const bool bit = sx[t] > 0.5f;
        #pragma unroll
        for (int j = 0; j < 4; ++j)
            h[j] = fmaf(bit ? a1[j] : a0[j], h[j], bit ? b1[j] : 0.0f);
        *(float4*)(tile + t * STRIDE + d0i) = make_float4(h[0], h[1], h[2], h[3]);
    }
    __syncthreads();

    // GEMV on the matrix pipe: each of the 4 waves owns 16 timesteps.
    // V_WMMA_F32_16X16X4_F32 layout (ISA 7.12.2): lane L, M = L&15,
    // A vgpr0/1 carry K = (L<16 ? 0,1 : 2,3); B mirrors with N = L&15.
    const int wave = tid >> 5, lane = tid & 31;
    const int m = lane & 15, hi = lane >> 4;
    const float* arow = tile + (wave * 16 + m) * STRIDE + 2 * hi;
    const float* brow = sC + 2 * hi;

    v8f acc0 = {0.f, 0.f, 0.f, 0.f, 0.f, 0.f, 0.f, 0.f};
    v8f acc1 = {0.f, 0.f, 0.f, 0.f, 0.f, 0.f, 0.f, 0.f};
    for (int k = 0; k < D_DIM; k += 8) {   // two accumulators hide WMMA latency
        const v2f av0 = *(const v2f*)(arow + k);
        const v2f bv0 = *(const v2f*)(brow + k);
        acc0 = __builtin_amdgcn_wmma_f32_16x16x4_f32(
            false, av0, false, bv0, (short)0, acc0, false, false);
        const v2f av1 = *(const v2f*)(arow + k + 4);
        const v2f bv1 = *(const v2f*)(brow + k + 4);
        acc1 = __builtin_amdgcn_wmma_f32_16x16x4_f32(
            false, av1, false, bv1, (short)0, acc1, false, false);
    }
    const v8f acc = acc0 + acc1;

    // Every output column N is identical (B replicated). Lane 0 of each wave
    // holds y for M=0..7 in acc[0..7]; lane 16 holds M=8..15.
    if (m == 0) {
        const float dbias = dscal[0];
        float o[8];
        #pragma unroll
        for (int j = 0; j < 8; ++j) {
            const float z = acc[j] + dbias;
            const float s = 1.0f / (1.0f + __expf(-z));
            o[j] = fminf(fmaxf(s, 1e-7f), 1.0f - 1e-7f);
        }
        const int tb = b * T_LEN + c * L_CHUNK + wave * 16 + hi * 8;
        *(float4*)(y + tb)     = make_float4(o[0], o[1], o[2], o[3]);
        *(float4*)(y + tb + 4) = make_float4(o[4], o[5], o[6], o[7]);
    }
}

extern "C" void kernel_launch(void* const* d_in, const int* in_sizes, int n_in,
                              void* d_out, int out_size, void* d_ws, size_t ws_size,
                              hipStream_t stream) {
    const float* x     = (const float*)d_in[0];
    const float* A_log = (const float*)d_in[1];
    const float* dw    = (const float*)d_in[2];
    const float* db    = (const float*)d_in[3];
    const float* Bw    = (const float*)d_in[4];
    const float* Bb    = (const float*)d_in[5];
    const float* Cvec  = (const float*)d_in[6];
    const float* dscal = (const float*)d_in[7];
    float* y = (float*)d_out;

    const size_t n_pq = (size_t)B_BATCH * NC * D_DIM;    // 1M floats each
    float* P  = (float*)d_ws;
    float* Q  = P + n_pq;
    float* H0 = Q + n_pq;                                // 12 MB total

    dim3 grid(NC, B_BATCH);                              // 2048 workgroups
    ssm_compose<<<grid, 128, 0, stream>>>(x, A_log, dw, db, Bw, Bb, P, Q);
    ssm_chunkscan<<<(B_BATCH * D_DIM) / 256, 256, 0, stream>>>(P, Q, H0);

    const size_t shbytes = (size_t)(L_CHUNK * STRIDE + L_CHUNK + D_DIM) * sizeof(float);
    ssm_replay<<<grid, 128, shbytes, stream>>>(x, A_log, dw, db, Bw, Bb,
                                               Cvec, dscal, H0, y);
}